// OnlineTripletLoss_19576460935202
// MI455X (gfx1250) — compile-verified
//
#include <hip/hip_runtime.h>
#include <hip/hip_bf16.h>

typedef float v2f __attribute__((ext_vector_type(2)));
typedef float v8f __attribute__((ext_vector_type(8)));

#define B 512
#define D 128
#define MARGIN 1.0f

// ---------------------------------------------------------------------------
// Kernel 1: per-row squared norms, and zero the (loss_sum, count) accumulators.
// ---------------------------------------------------------------------------
__global__ void tl_sqnorm_zero(const float* __restrict__ E,
                               float* __restrict__ sq,
                               float* __restrict__ accum) {
  int i = blockIdx.x * blockDim.x + threadIdx.x;
  if (i < 2) accum[i] = 0.0f;
  if (i < B) {
    const float* row = E + i * D;
    float s = 0.0f;
#pragma unroll 8
    for (int k = 0; k < D; ++k) {
      float v = row[k];
      s += v * v;
    }
    sq[i] = s;
  }
}

// ---------------------------------------------------------------------------
// Kernel 2: dist[i,j] = sq[i] + sq[j] - 2 * (E E^T)[i,j] via f32 WMMA tiles.
// One wave (32 lanes) per 16x16 output tile; K=128 swept in steps of 4 with
// V_WMMA_F32_16X16X4_F32.
//
// 32-bit A 16x4 layout : element(vgpr v, lane L): m = L&15, k = (L>>4)*2 + v
// 32-bit B 4x16 layout : element(vgpr v, lane L): n = L&15, k = (L>>4)*2 + v
// 32-bit C/D 16x16     : element(vgpr v, lane L): n = L&15, m = (L>>4)*8 + v
// ---------------------------------------------------------------------------
__global__ void tl_dist_wmma(const float* __restrict__ E,
                             const float* __restrict__ sq,
                             float* __restrict__ dist) {
  const int ti   = blockIdx.x * 16;   // tile row base (i)
  const int tj   = blockIdx.y * 16;   // tile col base (j)
  const int lane = threadIdx.x;       // 0..31, wave32, EXEC all ones
  const int half = lane >> 4;         // 0 or 1
  const int l15  = lane & 15;

  const float* __restrict__ Ai = E + (ti + l15) * D;  // A row m = l15
  const float* __restrict__ Bj = E + (tj + l15) * D;  // B col n = l15 (B = E^T)

  v8f acc = {};
#pragma unroll
  for (int k = 0; k < D; k += 4) {
    v2f a, b;
    const int k0 = k + half * 2;
    a.x = Ai[k0];
    a.y = Ai[k0 + 1];
    b.x = Bj[k0];
    b.y = Bj[k0 + 1];
    // 8 args: (neg_a, A, neg_b, B, c_mod, C, reuse_a, reuse_b)
    acc = __builtin_amdgcn_wmma_f32_16x16x4_f32(
        false, a, false, b, (short)0, acc, false, false);
  }

  // Epilogue: dist = sq[i] + sq[j] - 2*gram
  const int col = tj + l15;
  const float sqj = sq[col];
#pragma unroll
  for (int v = 0; v < 8; ++v) {
    const int row = ti + (half << 3) + v;
    dist[row * B + col] = sq[row] + sqj - 2.0f * acc[v];
  }
}

// ---------------------------------------------------------------------------
// Kernel 3: per-anchor triplet reduction. One block per anchor; the anchor's
// dist row and all labels live in LDS so the O(B^2) inner sweep is DS-only.
// accum[0] += sum over valid (p,n) of relu(d_ap - d_an + margin)
// accum[1] += numpos(a) * numneg(a)
// ---------------------------------------------------------------------------
__global__ void tl_triplet(const float* __restrict__ dist,
                           const int* __restrict__ labels,
                           float* __restrict__ accum) {
  __shared__ float drow[B];
  __shared__ int   lab[B];
  __shared__ float redf[256];
  __shared__ int   redp[256];
  __shared__ int   redn[256];

  const int a   = blockIdx.x;
  const int tid = threadIdx.x;

  for (int i = tid; i < B; i += 256) {
    drow[i] = dist[a * B + i];
    lab[i]  = labels[i];
  }
  __syncthreads();

  const int la = lab[a];

  int negc = 0;
  for (int i = tid; i < B; i += 256) negc += (lab[i] != la) ? 1 : 0;

  float acc  = 0.0f;
  int   posc = 0;
  for (int p = tid; p < B; p += 256) {
    if (p == a || lab[p] != la) continue;
    ++posc;
    const float dap_m = drow[p] + MARGIN;
#pragma unroll 4
    for (int n = 0; n < B; ++n) {
      if (lab[n] == la) continue;
      const float t = dap_m - drow[n];
      acc += (t > 0.0f) ? t : 0.0f;
    }
  }

  redf[tid] = acc;
  redp[tid] = posc;
  redn[tid] = negc;
  __syncthreads();
#pragma unroll
  for (int s = 128; s > 0; s >>= 1) {
    if (tid < s) {
      redf[tid] += redf[tid + s];
      redp[tid] += redp[tid + s];
      redn[tid] += redn[tid + s];
    }
    __syncthreads();
  }
  if (tid == 0) {
    atomicAdd(&accum[0], redf[0]);
    atomicAdd(&accum[1], (float)(redp[0] * redn[0]));
  }
}

// ---------------------------------------------------------------------------
// Kernel 4: finalize loss = loss_sum / count
// ---------------------------------------------------------------------------
__global__ void tl_finalize(const float* __restrict__ accum,
                            float* __restrict__ out) {
  if (threadIdx.x == 0 && blockIdx.x == 0) out[0] = accum[0] / accum[1];
}

extern "C" void kernel_launch(void* const* d_in, const int* in_sizes, int n_in,
                              void* d_out, int out_size, void* d_ws, size_t ws_size,
                              hipStream_t stream) {
  const float* E      = (const float*)d_in[0];  // [512,128] f32
  const int*   labels = (const int*)d_in[1];    // [512] i32
  float*       out    = (float*)d_out;          // scalar f32

  // Workspace layout: dist (512*512 f32) | sq (512 f32) | accum (2 f32)
  char* ws = (char*)d_ws;
  float* dist  = (float*)ws;
  float* sq    = (float*)(ws + (size_t)B * B * sizeof(float));
  float* accum = (float*)(ws + (size_t)B * B * sizeof(float) + B * sizeof(float));

  tl_sqnorm_zero<<<dim3((B + 255) / 256), dim3(256), 0, stream>>>(E, sq, accum);
  tl_dist_wmma<<<dim3(B / 16, B / 16), dim3(32), 0, stream>>>(E, sq, dist);
  tl_triplet<<<dim3(B), dim3(256), 0, stream>>>(dist, labels, accum);
  tl_finalize<<<dim3(1), dim3(1), 0, stream>>>(accum, out);
}